// MambaConvBlock_65060164599992
// MI455X (gfx1250) — compile-verified
//
#include <hip/hip_runtime.h>
#include <math.h>
#include <stdint.h>

// Problem constants (from the reference)
#define HH   3
#define NN   16384
#define DM   256
#define DS   16
#define DC   8
#define DI   512      // EXPAND * D_MODEL
#define DTR  16       // DT_RANK
#define XD   48       // DT_RANK + 2*D_STATE
// Chunked scan config: NC * LC == NN
#define NC   128
#define LC   128
// Conv time-tile
#define CT   64

typedef float v2f __attribute__((ext_vector_type(2)));
typedef float v8f __attribute__((ext_vector_type(8)));
typedef unsigned int v4u __attribute__((ext_vector_type(4)));
typedef int v4i __attribute__((ext_vector_type(4)));
typedef int v8i __attribute__((ext_vector_type(8)));

#if defined(__gfx1250__) && \
    __has_builtin(__builtin_amdgcn_tensor_load_to_lds) && \
    __has_builtin(__builtin_amdgcn_s_wait_tensorcnt)
#define USE_TDM 1
#else
#define USE_TDM 0
#endif

__device__ __forceinline__ v8f v8f_zero() {
  v8f z;
#pragma unroll
  for (int i = 0; i < 8; ++i) z[i] = 0.0f;
  return z;
}

__device__ __forceinline__ v8f wmma_f32_k4(v2f a, v2f b, v8f c) {
  // D = A(16x4,f32) * B(4x16,f32) + C(16x16,f32)
  return __builtin_amdgcn_wmma_f32_16x16x4_f32(false, a, false, b, (short)0, c,
                                               false, false);
}

__device__ __forceinline__ float silu_f(float v) {
  return v * (1.0f / (1.0f + __expf(-v)));
}
__device__ __forceinline__ float softplus_f(float v) {
  return (v > 20.0f) ? v : __logf(1.0f + __expf(v));
}

// Publish LDS that was written behind the compiler's back (TDM): the array
// address escapes into the asm, and the memory clobber means the asm may have
// stored through it -> subsequent LDS loads must be materialized.
__device__ __forceinline__ void lds_publish(float* p) {
  asm volatile("" : "+v"(p) :: "memory");
}

#if USE_TDM
// TDM 2-D tile load: copy a (height x width) fp32 tile with row stride
// `stride` (all in elements) from global `gaddr` into LDS byte offset
// `lds_off`. Descriptor per CDNA5 ISA (D# groups 0/1).
// 6-arg builtin form: (v4u g0, v8i g1, v4i g2, v4i g3, v8i g4, i32 cpol).
__device__ __forceinline__ void tdm_load_2d(const float* gaddr,
                                            unsigned lds_off, unsigned width,
                                            unsigned height, unsigned stride) {
  unsigned long long ga = (unsigned long long)(uintptr_t)gaddr;
  v4u g0 = {1u,                                    // count=1, user mode
            lds_off,                               // lds_addr (bytes)
            (unsigned)(ga & 0xFFFFFFFFu),          // global_addr[31:0]
            (unsigned)((ga >> 32) & 0x01FFFFFFu) | (2u << 30)};  // addr[56:32] | type=2
  v8i g1 = {(int)0x00020000u,                        // wg_mask=0, data_size=4B
            (int)((width & 0xFFFFu) << 16),          // tensor_dim0[15:0]
            (int)((width >> 16) | ((height & 0xFFFFu) << 16)),   // td0 hi | td1 lo
            (int)((height >> 16) | ((width & 0xFFFFu) << 16)),   // td1 hi | tile_dim0
            (int)(height & 0xFFFFu),                 // tile_dim1 | tile_dim2=0
            (int)stride,                             // tensor_dim0_stride[31:0]
            0, 0};                                   // stride hi, dim1_stride
  v4i gz4 = {0, 0, 0, 0};
  v8i gz8 = {0, 0, 0, 0, 0, 0, 0, 0};
  __builtin_amdgcn_tensor_load_to_lds(g0, g1, gz4, gz4, gz8, 0);
}
#endif

// ---------------------------------------------------------------------------
// Generic fp32 WMMA GEMM:  Y[m, n] = X[m, :] . W[n, :]   (per head)
//   KD  : reduction dim (also W row stride)
//   TPW : 16-col tiles per wave
//   MT  : 16-row tiles per wave (wave computes MT*16 x TPW*16)
//   EPI : 0 = plain store, 1 = split xz store (cols<DI -> Y0, else Y1),
//         2 = softplus(v + bias[col]) store
// blockDim = 128 (4 waves, consecutive M strips). Software-pipelined loads.
// ---------------------------------------------------------------------------
template <int KD, int TPW, int MT, int EPI>
__global__ __launch_bounds__(128) void wmma_gemm(
    const float* __restrict__ X, int ldx, size_t sX,
    const float* __restrict__ W, size_t sW,
    float* __restrict__ Y0, float* __restrict__ Y1, int ldy, size_t sY,
    const float* __restrict__ bias) {
  const int lane = threadIdx.x & 31;
  const int wave = threadIdx.x >> 5;
  const int l    = lane & 15;
  const int kb   = (lane >> 4) << 1;  // lanes 0-15 -> K pair {0,1}; 16-31 -> {2,3}
  const int hsel = lane >> 4;         // C/D row offset selector
  const int head = blockIdx.z;
  const int m0   = (blockIdx.x * 4 + wave) * (16 * MT);
  const int n0   = blockIdx.y * (TPW * 16);

  const float* Xh = X + (size_t)head * sX;
  const float* Wh = W + (size_t)head * sW;

  const float* ap[MT];
  const float* bp[TPW];
#pragma unroll
  for (int i = 0; i < MT; ++i)
    ap[i] = Xh + (size_t)(m0 + i * 16 + l) * ldx + kb;
#pragma unroll
  for (int j = 0; j < TPW; ++j)
    bp[j] = Wh + (size_t)(n0 + j * 16 + l) * KD + kb;

  v8f acc[MT][TPW];
#pragma unroll
  for (int i = 0; i < MT; ++i)
#pragma unroll
    for (int j = 0; j < TPW; ++j) acc[i][j] = v8f_zero();

  v2f a[MT], b[TPW];
#pragma unroll
  for (int i = 0; i < MT; ++i) a[i] = *reinterpret_cast<const v2f*>(ap[i]);
#pragma unroll
  for (int j = 0; j < TPW; ++j) b[j] = *reinterpret_cast<const v2f*>(bp[j]);

  // Main loop: prefetch k+4 fragments before issuing the k WMMAs.
#pragma unroll 2
  for (int k = 0; k + 4 < KD; k += 4) {
    v2f an[MT], bn[TPW];
#pragma unroll
    for (int i = 0; i < MT; ++i)
      an[i] = *reinterpret_cast<const v2f*>(ap[i] + k + 4);
#pragma unroll
    for (int j = 0; j < TPW; ++j)
      bn[j] = *reinterpret_cast<const v2f*>(bp[j] + k + 4);
#pragma unroll
    for (int i = 0; i < MT; ++i)
#pragma unroll
      for (int j = 0; j < TPW; ++j)
        acc[i][j] = wmma_f32_k4(a[i], b[j], acc[i][j]);
#pragma unroll
    for (int i = 0; i < MT; ++i) a[i] = an[i];
#pragma unroll
    for (int j = 0; j < TPW; ++j) b[j] = bn[j];
  }
  // Epilogue k-step.
#pragma unroll
  for (int i = 0; i < MT; ++i)
#pragma unroll
    for (int j = 0; j < TPW; ++j)
      acc[i][j] = wmma_f32_k4(a[i], b[j], acc[i][j]);

#pragma unroll
  for (int i = 0; i < MT; ++i) {
#pragma unroll
    for (int j = 0; j < TPW; ++j) {
      const int c = n0 + j * 16 + l;
      float bv = 0.0f;
      if (EPI == 2) bv = bias[head * DI + c];
#pragma unroll
      for (int r = 0; r < 8; ++r) {
        const int mm = m0 + i * 16 + r + hsel * 8;
        float v = acc[i][j][r];
        if (EPI == 0) {
          Y0[(size_t)head * sY + (size_t)mm * ldy + c] = v;
        } else if (EPI == 1) {
          if (c < DI) Y0[(size_t)head * sY + (size_t)mm * DI + c] = v;
          else        Y1[(size_t)head * sY + (size_t)mm * DI + (c - DI)] = v;
        } else {  // EPI == 2
          Y0[(size_t)head * sY + (size_t)mm * ldy + c] = softplus_f(v + bv);
        }
      }
    }
  }
}

// ---------------------------------------------------------------------------
// Depthwise causal conv (D_CONV=8) + SiLU with sliding register window:
// one thread per channel walks CT timesteps -> ~1 load per output.
// grid = (NN/CT, HH), block = 512 (one thread per channel d)
// ---------------------------------------------------------------------------
__global__ __launch_bounds__(512) void conv_silu_kernel(
    const float* __restrict__ x, const float* __restrict__ cw,
    const float* __restrict__ cb, float* __restrict__ u) {
  const int d    = threadIdx.x;
  const int t0   = blockIdx.x * CT;
  const int head = blockIdx.y;
  const float* xh = x + (size_t)head * NN * DI + d;
  float* uh       = u + (size_t)head * NN * DI + d;

  float w[DC];
  const float* wp = cw + ((size_t)head * DI + d) * DC;
#pragma unroll
  for (int k = 0; k < DC; ++k) w[k] = wp[k];
  const float bv = cb[head * DI + d];

  float win[DC];
#pragma unroll
  for (int k = 0; k < DC - 1; ++k) {
    int tt = t0 - (DC - 1) + k;
    win[k] = (tt >= 0) ? xh[(size_t)tt * DI] : 0.0f;
  }

  for (int t = 0; t < CT; ++t) {
    win[DC - 1] = xh[(size_t)(t0 + t) * DI];
    float acc = bv;
#pragma unroll
    for (int k = 0; k < DC; ++k) acc += w[k] * win[k];
#pragma unroll
    for (int k = 0; k < DC - 1; ++k) win[k] = win[k + 1];
    uh[(size_t)(t0 + t) * DI] = silu_f(acc);
  }
}

// ---------------------------------------------------------------------------
// Scan phase 1: per chunk, zero-init local scan -> final local state L,
// plus sum of delta over the chunk (chunk A-product = exp(A * sumdelta)).
// B rows staged into LDS via TDM (one descriptor per workgroup).
// ---------------------------------------------------------------------------
__global__ __launch_bounds__(512) void scan_phase1(
    const float* __restrict__ delta, const float* __restrict__ u,
    const float* __restrict__ xdbl, const float* __restrict__ A_log,
    float* __restrict__ Lst, float* __restrict__ sumd) {
  __shared__ float Bs[LC * DS];
  const int c = blockIdx.x, head = blockIdx.y, d = threadIdx.x;
  const int t0 = c * LC;

#if USE_TDM
  if (threadIdx.x < 32) {  // wave 0 issues the DMA: 128x16 tile, stride 48
    tdm_load_2d(xdbl + ((size_t)head * NN + t0) * XD + DTR, 0u, DS, LC, XD);
    __builtin_amdgcn_s_wait_tensorcnt(0);
  }
  lds_publish(Bs);  // TDM wrote Bs behind the compiler's back
#else
  for (int i = threadIdx.x; i < LC * DS; i += 512) {
    int tl = i >> 4, s = i & 15;
    Bs[i] = xdbl[((size_t)head * NN + t0 + tl) * XD + DTR + s];
  }
#endif
  __syncthreads();

  float A[DS], st[DS];
#pragma unroll
  for (int s = 0; s < DS; ++s) {
    A[s]  = -__expf(A_log[((size_t)head * DI + d) * DS + s]);
    st[s] = 0.0f;
  }
  float sd = 0.0f;
  for (int t = 0; t < LC; ++t) {
    size_t gi = ((size_t)head * NN + t0 + t) * DI + d;
    float dl = delta[gi];
    float du = dl * u[gi];
    sd += dl;
    const float* br = &Bs[t * DS];
#pragma unroll
    for (int s = 0; s < DS; ++s)
      st[s] = __expf(dl * A[s]) * st[s] + du * br[s];
  }
  size_t base = ((size_t)head * NC + c) * DI + d;
  sumd[base] = sd;
#pragma unroll
  for (int s = 0; s < DS; ++s) Lst[base * DS + s] = st[s];
}

// ---------------------------------------------------------------------------
// Scan phase 2: exclusive prefix over chunks, parallel over (head,d,s).
// ---------------------------------------------------------------------------
__global__ __launch_bounds__(256) void scan_phase2(
    const float* __restrict__ A_log, const float* __restrict__ sumd,
    const float* __restrict__ Lst, float* __restrict__ S0) {
  int gid = blockIdx.x * 256 + threadIdx.x;
  if (gid >= HH * DI * DS) return;
  int s = gid % DS, d = (gid / DS) % DI, head = gid / (DI * DS);
  float A = -__expf(A_log[((size_t)head * DI + d) * DS + s]);
  float S = 0.0f;
  for (int c = 0; c < NC; ++c) {
    size_t base = ((size_t)head * NC + c) * DI + d;
    S0[base * DS + s] = S;
    S = __expf(A * sumd[base]) * S + Lst[base * DS + s];
  }
}

// ---------------------------------------------------------------------------
// Scan phase 3: replay each chunk from its prefix state; emit
//   y = (sum_s state*C) + u*Dskip, gated by silu(z); write y over delta.
// B|C staged as one 128x32 TDM tile (B = cols 0..15, C = cols 16..31).
// ---------------------------------------------------------------------------
__global__ __launch_bounds__(512) void scan_phase3(
    float* __restrict__ delta_y, const float* __restrict__ u,
    const float* __restrict__ z, const float* __restrict__ xdbl,
    const float* __restrict__ A_log, const float* __restrict__ Dskip,
    const float* __restrict__ S0) {
  __shared__ float BC[LC * 2 * DS];
  const int c = blockIdx.x, head = blockIdx.y, d = threadIdx.x;
  const int t0 = c * LC;

#if USE_TDM
  if (threadIdx.x < 32) {  // wave 0 issues the DMA: 128x32 tile, stride 48
    tdm_load_2d(xdbl + ((size_t)head * NN + t0) * XD + DTR, 0u, 2 * DS, LC, XD);
    __builtin_amdgcn_s_wait_tensorcnt(0);
  }
  lds_publish(BC);  // TDM wrote BC behind the compiler's back
#else
  for (int i = threadIdx.x; i < LC * 2 * DS; i += 512) {
    int tl = i >> 5, s = i & 31;
    BC[i] = xdbl[((size_t)head * NN + t0 + tl) * XD + DTR + s];
  }
#endif
  __syncthreads();

  float A[DS], st[DS];
  size_t base = ((size_t)head * NC + c) * DI + d;
#pragma unroll
  for (int s = 0; s < DS; ++s) {
    A[s]  = -__expf(A_log[((size_t)head * DI + d) * DS + s]);
    st[s] = S0[base * DS + s];
  }
  float Dv = Dskip[head * DI + d];
  for (int t = 0; t < LC; ++t) {
    size_t gi = ((size_t)head * NN + t0 + t) * DI + d;
    float dl = delta_y[gi];
    float uu = u[gi];
    float du = dl * uu;
    const float* br = &BC[t * 2 * DS];
    const float* cr = br + DS;
    float y = 0.0f;
#pragma unroll
    for (int s = 0; s < DS; ++s) {
      st[s] = __expf(dl * A[s]) * st[s] + du * br[s];
      y += st[s] * cr[s];
    }
    y += uu * Dv;
    y *= silu_f(z[gi]);
    delta_y[gi] = y;
  }
}

// ---------------------------------------------------------------------------
extern "C" void kernel_launch(void* const* d_in, const int* in_sizes, int n_in,
                              void* d_out, int out_size, void* d_ws,
                              size_t ws_size, hipStream_t stream) {
  const float* h     = (const float*)d_in[0];
  const float* ipw   = (const float*)d_in[1];
  const float* cw    = (const float*)d_in[2];
  const float* cb    = (const float*)d_in[3];
  const float* xpw   = (const float*)d_in[4];
  const float* dtw   = (const float*)d_in[5];
  const float* dtb   = (const float*)d_in[6];
  const float* A_log = (const float*)d_in[7];
  const float* Dsk   = (const float*)d_in[8];
  const float* opw   = (const float*)d_in[9];
  float* out = (float*)d_out;

  const size_t nHD = (size_t)HH * NN * DI;
  float* xbuf = (float*)d_ws;                         // x -> delta -> y (reused)
  float* zbuf = xbuf + nHD;
  float* ubuf = zbuf + nHD;
  float* xdbl = ubuf + nHD;                           // H*N*48
  float* sumd = xdbl + (size_t)HH * NN * XD;          // H*NC*DI
  float* Lst  = sumd + (size_t)HH * NC * DI;          // H*NC*DI*DS
  float* S0   = Lst  + (size_t)HH * NC * DI * DS;     // H*NC*DI*DS

  dim3 blk(128);

  // K1: xz = h @ in_proj_w^T   -> split into x (xbuf) and z (zbuf)
  wmma_gemm<DM, 4, 2, 1><<<dim3(NN / 128, (2 * DI) / 64, HH), blk, 0, stream>>>(
      h, DM, (size_t)NN * DM, ipw, (size_t)(2 * DI) * DM,
      xbuf, zbuf, DI, (size_t)NN * DI, nullptr);

  // K2: u = silu(conv(x))
  conv_silu_kernel<<<dim3(NN / CT, HH), dim3(512), 0, stream>>>(xbuf, cw, cb,
                                                                ubuf);

  // K3: x_dbl = u @ x_proj_w^T  (48 cols: dt | B | C)
  wmma_gemm<DI, 3, 2, 0><<<dim3(NN / 128, 1, HH), blk, 0, stream>>>(
      ubuf, DI, (size_t)NN * DI, xpw, (size_t)XD * DI,
      xdbl, nullptr, XD, (size_t)NN * XD, nullptr);

  // K4: delta = softplus(dt @ dt_proj_w^T + dtb) -> overwrite xbuf (x is dead)
  wmma_gemm<DTR, 4, 2, 2><<<dim3(NN / 128, DI / 64, HH), blk, 0, stream>>>(
      xdbl, XD, (size_t)NN * XD, dtw, (size_t)DI * DTR,
      xbuf, nullptr, DI, (size_t)NN * DI, dtb);

  // K5-K7: chunked associative selective scan
  scan_phase1<<<dim3(NC, HH), dim3(512), 0, stream>>>(xbuf, ubuf, xdbl, A_log,
                                                      Lst, sumd);
  scan_phase2<<<dim3((HH * DI * DS + 255) / 256), dim3(256), 0, stream>>>(
      A_log, sumd, Lst, S0);
  scan_phase3<<<dim3(NC, HH), dim3(512), 0, stream>>>(xbuf, ubuf, zbuf, xdbl,
                                                      A_log, Dsk, S0);

  // K8: out = y @ out_proj_w^T
  wmma_gemm<DI, 4, 2, 0><<<dim3(NN / 128, DM / 64, HH), blk, 0, stream>>>(
      xbuf, DI, (size_t)NN * DI, opw, (size_t)DM * DI,
      out, nullptr, DM, (size_t)NN * DM, nullptr);
}